// CompositeLoss_91053306675239
// MI455X (gfx1250) — compile-verified
//
#include <hip/hip_runtime.h>

typedef __attribute__((ext_vector_type(2))) float v2f;
typedef __attribute__((ext_vector_type(4))) float fv4;
typedef __attribute__((ext_vector_type(8))) float v8f;

#define VOCAB  32000
#define VOCAB4 8000     // VOCAB / 4
#define NP     256      // predictions per image
#define NT     32       // targets per image
#define NC     80       // classes

// ---------------------------------------------------------------------------
// Exact f32 wave(32) sum via V_WMMA_F32_16X16X4_F32:
//   A (16x4): VGPR0 lanes0-15 -> A[m][0], lanes16-31 -> A[m][2]; VGPR1 -> K=1,3 (zeros)
//   B = ones(4x16)  =>  D[m][n] = sp[m] + sp[m+16]
//   D VGPR r: lanes0-15 hold row r, lanes16-31 hold row r+8.
//   sum_{r=0..7} d[r]  = (lanes<16) sum sp[0..7]+sp[16..23]
//                      = (lanes>=16) sum sp[8..15]+sp[24..31]
//   + shfl_xor(16)     = full 32-lane sum (all lanes).
// ---------------------------------------------------------------------------
__device__ __forceinline__ float wave_wmma_sum(float v) {
  v2f a; a[0] = v;    a[1] = 0.0f;
  v2f b; b[0] = 1.0f; b[1] = 1.0f;
  v8f c = {};
  v8f d = __builtin_amdgcn_wmma_f32_16x16x4_f32(false, a, false, b, (short)0, c,
                                                false, false);
  float t = d[0] + d[1] + d[2] + d[3] + d[4] + d[5] + d[6] + d[7];
  t += __shfl_xor(t, 16, 32);
  return t;
}

// ---------------------------------------------------------------------------
// LM cross-entropy: one block per (b,s) row, single streaming pass,
// online logsumexp per thread, nontemporal b128 loads (data > L2, read once).
// ---------------------------------------------------------------------------
__global__ __launch_bounds__(256) void lm_ce_kernel(
    const float* __restrict__ logits, const int* __restrict__ labels,
    float* __restrict__ ws) {
  const int row = blockIdx.x;
  const float* x = logits + (size_t)row * VOCAB;
  const fv4* x4 = (const fv4*)x;
  const int tid = threadIdx.x;

  float m = -__builtin_inff();
  float s = 0.0f;
  for (int k = tid; k < VOCAB4; k += 256) {
    fv4 v = __builtin_nontemporal_load(&x4[k]);
#pragma unroll
    for (int j = 0; j < 4; ++j) {
      float val = v[j];
      if (val > m) { s = s * __expf(m - val) + 1.0f; m = val; }
      else         { s += __expf(val - m); }
    }
  }

  // block max of per-thread maxima
  __shared__ float smax[8];
  __shared__ float ssum[8];
  const int wid = tid >> 5, lane = tid & 31;
  const float tm = m;
  float wm = tm;
#pragma unroll
  for (int off = 16; off > 0; off >>= 1) wm = fmaxf(wm, __shfl_xor(wm, off, 32));
  if (lane == 0) smax[wid] = wm;
  __syncthreads();
  float M = smax[0];
#pragma unroll
  for (int i = 1; i < 8; ++i) M = fmaxf(M, smax[i]);

  // rescale partial sums to common max, wave-sum them with WMMA
  const float sp = s * __expf(tm - M);
  const float S_w = wave_wmma_sum(sp);      // uniform flow: EXEC all ones
  if (lane == 0) ssum[wid] = S_w;
  __syncthreads();

  if (tid == 0) {
    float S = 0.0f;
    for (int i = 0; i < 8; ++i) S += ssum[i];
    const int lab = labels[row];
    if (lab != -100) {
      const float nll = (M + __logf(S)) - x[lab];
      atomicAdd(&ws[0], nll);   // nll sum
      atomicAdd(&ws[1], 1.0f);  // valid count
    }
  }
}

// ---------------------------------------------------------------------------
// Detection loss: one block per image. IoU matrix lives in LDS (32 KB).
// Greedy matching replicates jnp.argmax tie-breaking (first flat index).
// ---------------------------------------------------------------------------
__device__ __forceinline__ float sl1(float d) {
  return d < 1.0f ? 0.5f * d * d : d - 0.5f;
}

__global__ __launch_bounds__(256) void det_loss_kernel(
    const float* __restrict__ class_logits, const float* __restrict__ box_preds,
    const int* __restrict__ t_labels, const float* __restrict__ t_boxes,
    float* __restrict__ ws) {
  const int b = blockIdx.x;
  const int tid = threadIdx.x;

  __shared__ float pb[NP * 4];
  __shared__ float tb[NT * 4];
  __shared__ float iou[NP * NT];
  __shared__ float rv[256];
  __shared__ int   ri[256];
  __shared__ int   ps_s[NT];
  __shared__ int   ts_s[NT];
  __shared__ float vs_s[NT];

  for (int i = tid; i < NP * 4; i += 256) pb[i] = box_preds[(size_t)b * NP * 4 + i];
  if (tid < NT * 4) tb[tid] = t_boxes[(size_t)b * NT * 4 + tid];
  __syncthreads();

  // IoU: thread p handles row p
  {
    const int p = tid;
    const float ax1 = pb[p*4+0], ay1 = pb[p*4+1], ax2 = pb[p*4+2], ay2 = pb[p*4+3];
    const float area_a = (ax2 - ax1) * (ay2 - ay1);
    for (int t = 0; t < NT; ++t) {
      const float bx1 = tb[t*4+0], by1 = tb[t*4+1], bx2 = tb[t*4+2], by2 = tb[t*4+3];
      const float area_b = (bx2 - bx1) * (by2 - by1);
      const float lx = fmaxf(ax1, bx1), ly = fmaxf(ay1, by1);
      const float rx = fminf(ax2, bx2), ry = fminf(ay2, by2);
      const float w = fmaxf(rx - lx, 0.0f), h = fmaxf(ry - ly, 0.0f);
      const float inter = w * h;
      const float uni = area_a + area_b - inter;
      iou[p * NT + t] = inter / fmaxf(uni, 1e-7f);
    }
  }
  __syncthreads();

  // Greedy matching: 32 sequential global-argmax steps
  for (int step = 0; step < NT; ++step) {
    float bv = iou[tid * NT + 0];
    int   bi = tid * NT;
#pragma unroll
    for (int t = 1; t < NT; ++t) {
      const float v = iou[tid * NT + t];
      if (v > bv) { bv = v; bi = tid * NT + t; }   // strict > keeps first index
    }
    rv[tid] = bv; ri[tid] = bi;
    __syncthreads();
    for (int str = 128; str > 0; str >>= 1) {
      if (tid < str) {
        const float v2 = rv[tid + str];
        const int   i2 = ri[tid + str];
        if (v2 > rv[tid] || (v2 == rv[tid] && i2 < ri[tid])) { rv[tid] = v2; ri[tid] = i2; }
      }
      __syncthreads();
    }
    const int   bidx = ri[0];
    const float bval = rv[0];
    const int p = bidx / NT, t = bidx % NT;
    if (tid == 0) {
      ps_s[step] = p; ts_s[step] = t;
      vs_s[step] = (bval >= 0.5f) ? 1.0f : 0.0f;
    }
    // mask row p and column t (always, like the reference)
    if (tid < NT) iou[p * NT + tid] = -1.0f;
    iou[tid * NT + t] = -1.0f;
    __syncthreads();
  }

  // Matched-pair losses: one thread per match (wave 0)
  float local = 0.0f, vloc = 0.0f;
  if (tid < NT) {
    const int p = ps_s[tid], t = ts_s[tid];
    vloc = vs_s[tid];
    const float a0 = pb[p*4+0], a1 = pb[p*4+1], a2 = pb[p*4+2], a3 = pb[p*4+3];
    const float b0 = tb[t*4+0], b1 = tb[t*4+1], b2 = tb[t*4+2], b3 = tb[t*4+3];
    const float area_a = (a2 - a0) * (a3 - a1), area_b = (b2 - b0) * (b3 - b1);
    const float lx = fmaxf(a0, b0), ly = fmaxf(a1, b1);
    const float rx = fminf(a2, b2), ry = fminf(a3, b3);
    const float w = fmaxf(rx - lx, 0.0f), h = fmaxf(ry - ly, 0.0f);
    const float inter = w * h;
    const float uni = area_a + area_b - inter;
    const float iouv = inter / (uni + 1e-7f);
    const float elx = fminf(a0, b0), ely = fminf(a1, b1);
    const float erx = fmaxf(a2, b2), ery = fmaxf(a3, b3);
    const float ew = fmaxf(erx - elx, 0.0f), eh = fmaxf(ery - ely, 0.0f);
    const float ea = ew * eh;
    const float giou_l = 1.0f - (iouv - (ea - uni) / (ea + 1e-7f));
    const float l1 = 0.25f * (sl1(fabsf(a0 - b0)) + sl1(fabsf(a1 - b1)) +
                              sl1(fabsf(a2 - b2)) + sl1(fabsf(a3 - b3)));
    // class CE over 80 logits
    const float* cl = class_logits + ((size_t)b * NP + p) * NC;
    float cm = cl[0];
    for (int c2 = 1; c2 < NC; ++c2) cm = fmaxf(cm, cl[c2]);
    float cs = 0.0f;
    for (int c2 = 0; c2 < NC; ++c2) cs += __expf(cl[c2] - cm);
    const int tc = t_labels[b * NT + t];
    const float cls_l = -((cl[tc] - cm) - __logf(cs));
    local = (0.2f * cls_l + 0.8f * (0.7f * giou_l + 0.3f * l1)) * vloc;
  }
  if (tid < 32) {
    float mv = local, nv = vloc;
#pragma unroll
    for (int off = 16; off > 0; off >>= 1) {
      mv += __shfl_xor(mv, off, 32);
      nv += __shfl_xor(nv, off, 32);
    }
    if (tid == 0) {
      const float penalty = 0.5f * 0.8f * 0.3f + 0.5f * 0.2f;  // 0.22
      const float det = mv + (((float)NP - nv) + ((float)NT - nv)) * penalty;
      atomicAdd(&ws[2], det);
    }
  }
}

__global__ void finalize_kernel(const float* __restrict__ ws, float* __restrict__ out) {
  const float cnt = fmaxf(ws[1], 1.0f);
  out[0] = 0.1f * (ws[0] / cnt) + ws[2];
}

// ---------------------------------------------------------------------------
extern "C" void kernel_launch(void* const* d_in, const int* in_sizes, int n_in,
                              void* d_out, int out_size, void* d_ws, size_t ws_size,
                              hipStream_t stream) {
  const float* lm_logits     = (const float*)d_in[0];
  const int*   lm_labels     = (const int*)d_in[1];
  const float* class_logits  = (const float*)d_in[2];
  const float* box_preds     = (const float*)d_in[3];
  const int*   target_labels = (const int*)d_in[4];
  const float* target_boxes  = (const float*)d_in[5];
  float* out = (float*)d_out;
  float* ws  = (float*)d_ws;

  const int rows = in_sizes[1];        // B*S = 4096
  const int nimg = in_sizes[4] / NT;   // B = 4

  hipMemsetAsync(ws, 0, 4 * sizeof(float), stream);  // graph-capturable
  lm_ce_kernel<<<rows, 256, 0, stream>>>(lm_logits, lm_labels, ws);
  det_loss_kernel<<<nimg, 256, 0, stream>>>(class_logits, box_preds,
                                            target_labels, target_boxes, ws);
  finalize_kernel<<<1, 1, 0, stream>>>(ws, out);
}